// TestSoftNMSModule_3040836846183
// MI455X (gfx1250) — compile-verified
//
#include <hip/hip_runtime.h>
#include <hip/hip_bf16.h>
#include <math.h>

// Soft-NMS (sigma=0.5, thresh=0.05, N=4096) for MI455X / gfx1250.
//
// Latency-bound sequential algorithm: single persistent workgroup (32 wave32
// waves on one WGP), full working set resident in LDS (CDNA5: 320KB/WGP),
// boxes staged via the gfx1250 async global->LDS path (ASYNCcnt), per-step
// argmax via wave32 shfl_xor reductions.

#define NBOX     4096
#define NTHR     1024
#define EPT      (NBOX / NTHR)   // 4 boxes per thread
#define THRESH_F 0.05f           // SCORE_THRESH
// SIGMA = 0.5 -> w = exp(-iou^2/0.5) = exp(-2*iou^2)

__global__ __launch_bounds__(NTHR, 1)
void soft_nms_gfx1250(const float* __restrict__ boxes,
                      const float* __restrict__ scores,
                      float* __restrict__ out)
{
    __shared__ float s_box[NBOX * 4];   // 64 KB, AoS x1,y1,x2,y2
    __shared__ float s_area[NBOX];      // 16 KB
    __shared__ float s_avail[NBOX];     // 16 KB; -inf == processed
    __shared__ float s_rv[32];
    __shared__ int   s_ri[32];
    __shared__ float s_sel[8];          // [0..3] box, [4] score, [5] area
    __shared__ int   s_isel;

    const int tid = (int)threadIdx.x;

    // ---- Stage boxes (64 KB) into LDS via CDNA5 async global->LDS DMA ----
    // Each thread issues EPT b128 async copies; tracked by ASYNCcnt.
    #pragma unroll
    for (int k = 0; k < EPT; ++k) {
        const int c = tid + k * NTHR;  // 16-byte chunk index
        unsigned lds_addr = (unsigned)(uintptr_t)(&s_box[c * 4]); // low 32b = LDS offset
        unsigned long long gaddr = (unsigned long long)(uintptr_t)(boxes + c * 4);
        asm volatile("global_load_async_to_lds_b128 %0, %1, off"
                     :: "v"(lds_addr), "v"(gaddr) : "memory");
    }
    // Scores via the regular vector path (overlaps with the async DMA);
    // also zero the final-score half of the output.
    #pragma unroll
    for (int k = 0; k < EPT; ++k) {
        const int j = tid + k * NTHR;
        s_avail[j] = scores[j];
        out[j] = 0.0f;
    }
    asm volatile("s_wait_asynccnt 0" ::: "memory");
    __syncthreads();

    // Areas from LDS-resident boxes.
    #pragma unroll
    for (int k = 0; k < EPT; ++k) {
        const int j = tid + k * NTHR;
        const float x1 = s_box[4 * j + 0], y1 = s_box[4 * j + 1];
        const float x2 = s_box[4 * j + 2], y2 = s_box[4 * j + 3];
        s_area[j] = (x2 - x1) * (y2 - y1);
    }
    __syncthreads();

    for (int t = 0; t < NBOX; ++t) {
        // ---- argmax over s_avail (first-occurrence tie-break, like jnp.argmax)
        float bv = s_avail[tid];
        int   bi = tid;
        #pragma unroll
        for (int k = 1; k < EPT; ++k) {
            const int j = tid + k * NTHR;
            const float v = s_avail[j];
            if (v > bv) { bv = v; bi = j; }   // strict '>' keeps lowest index
        }
        // wave32 butterfly reduction
        #pragma unroll
        for (int m = 16; m > 0; m >>= 1) {
            const float ov = __shfl_xor(bv, m, 32);
            const int   oi = __shfl_xor(bi, m, 32);
            if (ov > bv || (ov == bv && oi < bi)) { bv = ov; bi = oi; }
        }
        if ((tid & 31) == 0) { s_rv[tid >> 5] = bv; s_ri[tid >> 5] = bi; }
        __syncthreads();
        if (tid < 32) {
            bv = s_rv[tid]; bi = s_ri[tid];
            #pragma unroll
            for (int m = 16; m > 0; m >>= 1) {
                const float ov = __shfl_xor(bv, m, 32);
                const int   oi = __shfl_xor(bi, m, 32);
                if (ov > bv || (ov == bv && oi < bi)) { bv = ov; bi = oi; }
            }
            if (tid == 0) {
                s_isel  = bi;
                s_sel[0] = s_box[4 * bi + 0];
                s_sel[1] = s_box[4 * bi + 1];
                s_sel[2] = s_box[4 * bi + 2];
                s_sel[3] = s_box[4 * bi + 3];
                s_sel[4] = bv;
                s_sel[5] = s_area[bi];
                out[NBOX + t] = (float)bi;           // idxs output (as float)
                if (bv > THRESH_F) {
                    out[bi] = bv;                    // final score scattered once
                    s_avail[bi] = -__builtin_inff(); // mark processed
                }
            }
        }
        __syncthreads();

        const float sel = s_sel[4];
        if (!(sel > THRESH_F)) {
            // Fixed point reached: every remaining scan step picks the same
            // index with out_score 0. Fill idxs and exit.
            const float fi = (float)s_isel;
            for (int j = t + 1 + tid; j < NBOX; j += NTHR)
                out[NBOX + j] = fi;
            break;
        }

        // ---- IoU of winner vs all boxes; Gaussian decay of live scores.
        const float bx1 = s_sel[0], by1 = s_sel[1];
        const float bx2 = s_sel[2], by2 = s_sel[3];
        const float ba  = s_sel[5];
        #pragma unroll
        for (int k = 0; k < EPT; ++k) {
            const int j = tid + k * NTHR;
            const float x1 = fmaxf(bx1, s_box[4 * j + 0]);
            const float y1 = fmaxf(by1, s_box[4 * j + 1]);
            const float x2 = fminf(bx2, s_box[4 * j + 2]);
            const float y2 = fminf(by2, s_box[4 * j + 3]);
            const float inter = fmaxf(x2 - x1, 0.0f) * fmaxf(y2 - y1, 0.0f);
            const float iou = inter / (ba + s_area[j] - inter);
            const float w = expf(iou * iou * -2.0f);   // exp(-iou^2/sigma), sigma=0.5
            s_avail[j] *= w;   // -inf stays -inf for processed entries
        }
        __syncthreads();
    }
}

extern "C" void kernel_launch(void* const* d_in, const int* in_sizes, int n_in,
                              void* d_out, int out_size, void* d_ws, size_t ws_size,
                              hipStream_t stream) {
    (void)in_sizes; (void)n_in; (void)out_size; (void)d_ws; (void)ws_size;
    const float* boxes  = (const float*)d_in[0];   // [N,4] f32
    const float* scores = (const float*)d_in[1];   // [N]   f32
    float* out = (float*)d_out;                    // [2N]: final_scores | idxs
    hipLaunchKernelGGL(soft_nms_gfx1250, dim3(1), dim3(NTHR), 0, stream,
                       boxes, scores, out);
}